// ParamLSTM_44495861186788
// MI455X (gfx1250) — compile-verified
//
#include <hip/hip_runtime.h>
#include <math.h>
#include <stdint.h>

typedef float v2f __attribute__((ext_vector_type(2)));
typedef float v8f __attribute__((ext_vector_type(8)));

#define Bn 64
#define In 512
#define Hn 1024

#define KCH   16                    // K-chunk per stage
#define PITCH 20                    // dwords per staged row (16 + 4 pad); bank-conflict-free
#define REGION_DW (4 * 16 * PITCH)  // 4 gates x 16 rows x PITCH = 1280 dwords per region
#define NREG  3                     // triple-buffered pipeline (depth 2 chunks)
#define WAVES 4                     // waves per block
#define ROWS_PER_BLOCK (WAVES * 16) // 64 rows

// CDNA5 async global->LDS copy, GVS mode: mem = SADDR + VADDR32 ; lds = VDST32.
#define ASYNC_B128(loff, voff, sbase)                                      \
    asm volatile("global_load_async_to_lds_b128 %0, %1, %2"                \
                 :: "v"(loff), "v"(voff), "s"(sbase) : "memory")
#define WAIT_ASYNCCNT(n)  asm volatile("s_wait_asynccnt " #n ::: "memory")
#define WAIT_DSCNT0()     asm volatile("s_wait_dscnt 0x0" ::: "memory")

__device__ __forceinline__ float sigmoidf_(float v) { return 1.0f / (1.0f + __expf(-v)); }

// Issue 8 async B128 copies: 4 gates x (16 rows x KCH) fp32 tile -> LDS region.
// Addressing: SGPR base (gate ptr + k0*4, uniform) + constant per-lane VGPR offset.
__device__ __forceinline__ void stage_async(
    const uint64_t gb[4], int k0, const float* region,
    const uint32_t ldsC[8], uint32_t voff0, uint32_t voff1)
{
    WAIT_DSCNT0();  // async LDS writes are not ordered with DS ops: close WAR race
    const uint32_t rB = (uint32_t)(uintptr_t)region;   // LDS aperture: addr[31:0]
    #pragma unroll
    for (int g = 0; g < 4; ++g) {
        const uint64_t sb = gb[g] + (uint64_t)(uint32_t)(k0 * 4);
        ASYNC_B128(rB + ldsC[2 * g + 0], voff0, sb);   // rows 0..7
        ASYNC_B128(rB + ldsC[2 * g + 1], voff1, sb);   // rows 8..15
    }
}

// 4 WMMA K-steps per gate: D(16x16) = W(16x4) x v(4x16 broadcast) + C
__device__ __forceinline__ void compute_chunk(
    const float* region, const float* vec, int k0,
    int laneRow, int laneHi, v8f& a0, v8f& a1, v8f& a2, v8f& a3)
{
    #pragma unroll
    for (int kk = 0; kk < KCH; kk += 4) {
        v2f bv = *(const v2f*)(vec + k0 + kk + 2 * laneHi);
        v2f m0 = *(const v2f*)&region[(0 * 16 + laneRow) * PITCH + kk + 2 * laneHi];
        v2f m1 = *(const v2f*)&region[(1 * 16 + laneRow) * PITCH + kk + 2 * laneHi];
        v2f m2 = *(const v2f*)&region[(2 * 16 + laneRow) * PITCH + kk + 2 * laneHi];
        v2f m3 = *(const v2f*)&region[(3 * 16 + laneRow) * PITCH + kk + 2 * laneHi];
        a0 = __builtin_amdgcn_wmma_f32_16x16x4_f32(false, m0, false, bv, (short)0, a0, false, false);
        a1 = __builtin_amdgcn_wmma_f32_16x16x4_f32(false, m1, false, bv, (short)0, a1, false, false);
        a2 = __builtin_amdgcn_wmma_f32_16x16x4_f32(false, m2, false, bv, (short)0, a2, false, false);
        a3 = __builtin_amdgcn_wmma_f32_16x16x4_f32(false, m3, false, bv, (short)0, a3, false, false);
    }
}

__global__ __launch_bounds__(WAVES * 32) void lstm_fused_wmma(
    const float* __restrict__ xin,
    const float* __restrict__ h0,
    const float* __restrict__ c0,
    const float* __restrict__ w_hi, const float* __restrict__ w_xi,
    const float* __restrict__ w_hf, const float* __restrict__ w_xf,
    const float* __restrict__ w_hg, const float* __restrict__ w_xg,
    const float* __restrict__ w_ho, const float* __restrict__ w_xo,
    const float* __restrict__ b_hi, const float* __restrict__ b_xi,
    const float* __restrict__ b_hf, const float* __restrict__ b_xf,
    const float* __restrict__ b_hg, const float* __restrict__ b_xg,
    const float* __restrict__ b_ho, const float* __restrict__ b_xo,
    float* __restrict__ out)          // [x_out | h_state | c_state]
{
    __shared__ float lds[WAVES * NREG * REGION_DW];   // 60 KB: 3 regions per wave

    const int tid  = threadIdx.x;
    const int wave = tid >> 5;
    const int lane = tid & 31;
    const int b    = blockIdx.x & (Bn - 1);        // grid = Bn * (Hn/64)
    const int tile = blockIdx.x >> 6;              // 0..15
    const int r0   = tile * ROWS_PER_BLOCK + wave * 16;

    float* R0 = &lds[wave * NREG * REGION_DW];
    float* R1 = R0 + REGION_DW;
    float* R2 = R1 + REGION_DW;

    const int laneRow = lane & 15;       // WMMA A/D row within half-wave
    const int laneHi  = lane >> 4;       // 0: K=0..1 / M=j ; 1: K=2..3 / M=j+8
    const int crow    = lane >> 2;       // staging: row group 0..7
    const int ccolDW  = (lane & 3) << 2; // staging: dword column 0,4,8,12

    // per-lane LDS byte offsets (region-relative) for the 8 staging targets
    uint32_t ldsC[8];
    #pragma unroll
    for (int g = 0; g < 4; ++g)
        #pragma unroll
        for (int half = 0; half < 2; ++half)
            ldsC[2 * g + half] =
                (uint32_t)(((g * 16 + half * 8 + crow) * PITCH + ccolDW) * 4);

    v8f a0 = {}, a1 = {}, a2 = {}, a3 = {};

    #pragma unroll
    for (int phase = 0; phase < 2; ++phase) {
        const int K = phase ? In : Hn;
        const float* vec = phase ? (xin + (size_t)b * In) : (h0 + (size_t)b * Hn);
        const size_t wofs = (size_t)b * Hn * K + (size_t)r0 * K;
        const uint64_t gb[4] = {
            (uint64_t)(uintptr_t)((phase ? w_xi : w_hi) + wofs),
            (uint64_t)(uintptr_t)((phase ? w_xf : w_hf) + wofs),
            (uint64_t)(uintptr_t)((phase ? w_xg : w_hg) + wofs),
            (uint64_t)(uintptr_t)((phase ? w_xo : w_ho) + wofs)
        };
        // per-lane global byte offsets (constant per phase): row*K + col
        const uint32_t voff0 = (uint32_t)(((crow)     * K + ccolDW) * 4);
        const uint32_t voff1 = (uint32_t)(((8 + crow) * K + ccolDW) * 4);

        const int NC = K / KCH;          // 64 or 32 chunks

        // pipeline prologue: fill two regions ahead
        stage_async(gb, 0 * KCH, R0, ldsC, voff0, voff1);
        stage_async(gb, 1 * KCH, R1, ldsC, voff0, voff1);

        float* Ra = R0; float* Rb = R1; float* Rc = R2;
        for (int n = 0; n + 2 < NC; ++n) {
            stage_async(gb, (n + 2) * KCH, Rc, ldsC, voff0, voff1);
            WAIT_ASYNCCNT(0x10);         // <=16 outstanding -> oldest 8 (chunk n) done
            compute_chunk(Ra, vec, n * KCH, laneRow, laneHi, a0, a1, a2, a3);
            float* t = Ra; Ra = Rb; Rb = Rc; Rc = t;
        }
        WAIT_ASYNCCNT(0x8);              // chunk NC-2 done
        compute_chunk(Ra, vec, (NC - 2) * KCH, laneRow, laneHi, a0, a1, a2, a3);
        WAIT_ASYNCCNT(0x0);              // chunk NC-1 done
        compute_chunk(Rb, vec, (NC - 1) * KCH, laneRow, laneHi, a0, a1, a2, a3);
    }

    // epilogue: lane 0 holds y[0..7], lane 16 holds y[8..15] (dup across N)
    if (laneRow == 0) {
        const int rbase = r0 + 8 * laneHi;
        #pragma unroll
        for (int j = 0; j < 8; ++j) {
            const size_t idx = (size_t)b * Hn + (size_t)(rbase + j);
            float gi = a0[j] + b_hi[idx] + b_xi[idx];
            float gf = a1[j] + b_hf[idx] + b_xf[idx];
            float gg = a2[j] + b_hg[idx] + b_xg[idx];
            float go = a3[j] + b_ho[idx] + b_xo[idx];
            float it = sigmoidf_(gi);
            float ft = sigmoidf_(gf);
            float gt = tanhf(gg);
            float ot = sigmoidf_(go);
            float cs = ft * c0[idx] + it * gt;
            float hs = ot * tanhf(cs);
            out[idx] = hs;                              // x_out (B,H)
            out[(size_t)Bn * Hn + idx] = hs;            // h_state (B,H,1)
            out[2ull * (size_t)Bn * Hn + idx] = cs;     // c_state (B,H,1)
        }
    }
}

extern "C" void kernel_launch(void* const* d_in, const int* in_sizes, int n_in,
                              void* d_out, int out_size, void* d_ws, size_t ws_size,
                              hipStream_t stream) {
    (void)in_sizes; (void)n_in; (void)out_size; (void)d_ws; (void)ws_size;
    const float* xin  = (const float*)d_in[0];
    const float* h0   = (const float*)d_in[1];
    const float* c0   = (const float*)d_in[2];
    const float* w_hi = (const float*)d_in[3];
    const float* w_xi = (const float*)d_in[4];
    const float* w_hf = (const float*)d_in[5];
    const float* w_xf = (const float*)d_in[6];
    const float* w_hg = (const float*)d_in[7];
    const float* w_xg = (const float*)d_in[8];
    const float* w_ho = (const float*)d_in[9];
    const float* w_xo = (const float*)d_in[10];
    const float* b_hi = (const float*)d_in[11];
    const float* b_xi = (const float*)d_in[12];
    const float* b_hf = (const float*)d_in[13];
    const float* b_xf = (const float*)d_in[14];
    const float* b_hg = (const float*)d_in[15];
    const float* b_xg = (const float*)d_in[16];
    const float* b_ho = (const float*)d_in[17];
    const float* b_xo = (const float*)d_in[18];
    float* out = (float*)d_out;

    dim3 grid(Bn * (Hn / ROWS_PER_BLOCK));   // 64 x 16 = 1024 blocks
    lstm_fused_wmma<<<grid, dim3(WAVES * 32), 0, stream>>>(
        xin, h0, c0,
        w_hi, w_xi, w_hf, w_xf, w_hg, w_xg, w_ho, w_xo,
        b_hi, b_xi, b_hf, b_xf, b_hg, b_xg, b_ho, b_xo,
        out);
}